// Atlas_31808527794849
// MI455X (gfx1250) — compile-verified
//
#include <hip/hip_runtime.h>

#define HO 256
#define WO 256
#define NB 2
#define PP 24
#define NCH 16

typedef float vf2 __attribute__((ext_vector_type(2)));
typedef float vf4 __attribute__((ext_vector_type(4)));

__device__ __forceinline__ int iclamp(int v, int lo, int hi) {
    return v < lo ? lo : (v > hi ? hi : v);
}

// ---------------------------------------------------------------------------
// Transpose one level from [P][N][H][W] to [P][H][W][N] (channel-last) so the
// gather kernel can fetch all 16 channels of a texel with 4 contiguous
// global_load_b128 (one L2 line) instead of 16 strided 4B gathers.
// Source reads use TH=NT (read-once stream) so the channel-last workspace —
// not the dead original layout — stays resident in the 192MB L2.
// Block = 128 threads handles a (16 n) x (32 x) tile for one (p, y).
// LDS tile padded [32][17] -> conflict-free scatter on the load phase.
// ---------------------------------------------------------------------------
__global__ __launch_bounds__(128) void atlas_transpose(
    const float* __restrict__ src, float* __restrict__ dst, int H, int W) {
    __shared__ float tile[32][17];  // [x][n], padded
    const int t  = threadIdx.x;
    const int x0 = blockIdx.x << 5;
    const int y  = blockIdx.y;
    const int p  = blockIdx.z;
    const size_t HW = (size_t)H * W;
    const float* s = src + (size_t)p * NCH * HW + (size_t)y * W + x0;
#pragma unroll
    for (int c = 0; c < 4; ++c) {
        int e = t + (c << 7);     // 0..511
        int n = e >> 5;           // 0..15
        int x = e & 31;           // 0..31
        tile[x][n] = __builtin_nontemporal_load(&s[(size_t)n * HW + x]);
    }
    __syncthreads();
    const int x  = t >> 2;
    const int n0 = (t & 3) << 2;
    vf4 v = {tile[x][n0], tile[x][n0 + 1], tile[x][n0 + 2], tile[x][n0 + 3]};
    float* dp = dst + (((size_t)p * H + y) * W + x0 + x) * NCH + n0;
    *(vf4*)dp = v;                // fully coalesced 16B/lane stores, RT (hot)
}

// ---------------------------------------------------------------------------
// Gather kernel: one thread per output pixel (b, ho, wo); 16 f32 accumulators.
// TR=true  : textures are channel-last [P][H][W][N] in workspace (fast path)
// TR=false : original [P][N][H][W] layout (fallback if ws too small)
// iuv loads and output stores are nontemporal (pure streams); texel gathers
// use default RT so they live in L2.
// ---------------------------------------------------------------------------
template <bool TR>
__global__ __launch_bounds__(256) void atlas_gather(
    const vf2* __restrict__ iuv,                          // [B][P][HO*WO]
    const float* __restrict__ t0, const float* __restrict__ t1,
    const float* __restrict__ t2, const float* __restrict__ t3,
    float* __restrict__ out) {                            // [B][16][HO*WO]
    const int tid = blockIdx.x * 256 + threadIdx.x;
    const int pix = tid & (HO * WO - 1);
    const int b   = tid >> 16;   // HO*WO == 65536

    float acc[NCH];
#pragma unroll
    for (int n = 0; n < NCH; ++n) acc[n] = 0.0f;

    const float* tex[4] = {t0, t1, t2, t3};
    const vf2* gbase = iuv + (size_t)b * PP * (HO * WO) + pix;

    for (int p = 0; p < PP; ++p) {
        vf2 g = __builtin_nontemporal_load(&gbase[(size_t)p * (HO * WO)]);
        // gfx1250 global_prefetch_b8 for the next grid line (clamped, branchless)
        int pn = p + 1 < PP ? p + 1 : p;
        __builtin_prefetch(&gbase[(size_t)pn * (HO * WO)], 0, 1);

#pragma unroll
        for (int l = 0; l < 4; ++l) {
            const int   hw    = 256 >> l;
            const float scale = 0.5f * (float)(hw - 1);
            // exact reference math: x = (g+1) * 0.5 * (w-1), align_corners
            float xf  = (g.x + 1.0f) * scale;
            float yf  = (g.y + 1.0f) * scale;
            float x0f = floorf(xf);
            float y0f = floorf(yf);
            float wx1 = xf - x0f;
            float wy1 = yf - y0f;
            float wx0 = 1.0f - wx1;
            float wy0 = 1.0f - wy1;
            // xf,yf in [0, hw-1] by construction; the only out-of-range corner
            // (x0+1 == hw when xf == hw-1) carries weight exactly 0, so
            // clamped indexing reproduces the reference's zero-padding.
            int ix0 = iclamp((int)x0f,     0, hw - 1);
            int iy0 = iclamp((int)y0f,     0, hw - 1);
            int ix1 = iclamp((int)x0f + 1, 0, hw - 1);
            int iy1 = iclamp((int)y0f + 1, 0, hw - 1);
            float w00 = wy0 * wx0, w01 = wy0 * wx1;
            float w10 = wy1 * wx0, w11 = wy1 * wx1;

            if (TR) {
                const vf4* t4 = (const vf4*)tex[l];
                const int pb = p * hw * hw;  // texel index base for this part
                const int idx[4] = {(pb + iy0 * hw + ix0) * 4,
                                    (pb + iy0 * hw + ix1) * 4,
                                    (pb + iy1 * hw + ix0) * 4,
                                    (pb + iy1 * hw + ix1) * 4};
                const float wgt[4] = {w00, w01, w10, w11};
#pragma unroll
                for (int c = 0; c < 4; ++c) {
                    const float w = wgt[c];
#pragma unroll
                    for (int q = 0; q < 4; ++q) {
                        vf4 v = t4[idx[c] + q];   // global_load_b128, RT
                        acc[q * 4 + 0] = fmaf(v.x, w, acc[q * 4 + 0]);
                        acc[q * 4 + 1] = fmaf(v.y, w, acc[q * 4 + 1]);
                        acc[q * 4 + 2] = fmaf(v.z, w, acc[q * 4 + 2]);
                        acc[q * 4 + 3] = fmaf(v.w, w, acc[q * 4 + 3]);
                    }
                }
            } else {
                const float* tl = tex[l];
                const size_t hw2 = (size_t)hw * hw;
                const size_t pb  = (size_t)p * NCH * hw2;
                const size_t offs[4] = {pb + (size_t)iy0 * hw + ix0,
                                        pb + (size_t)iy0 * hw + ix1,
                                        pb + (size_t)iy1 * hw + ix0,
                                        pb + (size_t)iy1 * hw + ix1};
                const float wgt[4] = {w00, w01, w10, w11};
#pragma unroll
                for (int c = 0; c < 4; ++c) {
                    const float w = wgt[c];
#pragma unroll
                    for (int n = 0; n < NCH; ++n)
                        acc[n] = fmaf(tl[offs[c] + (size_t)n * hw2], w, acc[n]);
                }
            }
        }
    }

#pragma unroll
    for (int n = 0; n < NCH; ++n)
        __builtin_nontemporal_store(
            acc[n], &out[((size_t)b * NCH + n) * (HO * WO) + pix]);
}

extern "C" void kernel_launch(void* const* d_in, const int* in_sizes, int n_in,
                              void* d_out, int out_size, void* d_ws, size_t ws_size,
                              hipStream_t stream) {
    (void)in_sizes; (void)n_in; (void)out_size;
    const vf2* iuv  = (const vf2*)d_in[0];
    const float* L1 = (const float*)d_in[1];   // [24][16][256][256]
    const float* L2 = (const float*)d_in[2];   // [24][16][128][128]
    const float* L3 = (const float*)d_in[3];   // [24][16][ 64][ 64]
    const float* L4 = (const float*)d_in[4];   // [24][16][ 32][ 32]
    float* out = (float*)d_out;

    const size_t n1 = (size_t)PP * NCH * 256 * 256;   // 25,165,824 floats
    const size_t n2 = n1 / 4, n3 = n1 / 16, n4 = n1 / 64;
    const size_t need = (n1 + n2 + n3 + n4) * sizeof(float);  // ~127.5 MB

    const int npix   = NB * HO * WO;          // 131072, multiple of 256
    const int blocks = npix / 256;

    if (ws_size >= need) {
        float* w1 = (float*)d_ws;
        float* w2 = w1 + n1;
        float* w3 = w2 + n2;
        float* w4 = w3 + n3;
        atlas_transpose<<<dim3(256 / 32, 256, PP), 128, 0, stream>>>(L1, w1, 256, 256);
        atlas_transpose<<<dim3(128 / 32, 128, PP), 128, 0, stream>>>(L2, w2, 128, 128);
        atlas_transpose<<<dim3( 64 / 32,  64, PP), 128, 0, stream>>>(L3, w3,  64,  64);
        atlas_transpose<<<dim3( 32 / 32,  32, PP), 128, 0, stream>>>(L4, w4,  32,  32);
        atlas_gather<true><<<blocks, 256, 0, stream>>>(iuv, w1, w2, w3, w4, out);
    } else {
        atlas_gather<false><<<blocks, 256, 0, stream>>>(iuv, L1, L2, L3, L4, out);
    }
}